// SGL_10780367913786
// MI455X (gfx1250) — compile-verified
//
#include <hip/hip_runtime.h>
#include <hip/hip_bf16.h>
#include <stdint.h>

// ---------------------------------------------------------------------------
// LightGCN-style propagation for MI455X (gfx1250, wave32).
//
// 9 COO SpMMs over an L2-resident working set (25.6MB embeddings + 15MB edges
// << 192MB L2).  Memory-bound on L2/atomic throughput, not HBM (23.3TB/s HBM
// would cover the one-time ~110MB read + 77MB write in ~10us; the real cost
// is 720M global_atomic_add_f32 ops and 2.9GB of L2 gather traffic).
//   - SpMM: 16 lanes/edge, float4 gather of x[col], 4x global_atomic_add_f32
//     scatter into y[row] (unsafeAtomicAdd -> hardware f32 atomic, no CAS).
//   - Edge triples double-buffered into LDS via async global->LDS DMA
//     (global_load_async_to_lds_b32 + s_wait_asynccnt), CDNA5 async path.
//   - Final 0.25 scale of the main view via v_wmma_f32_16x16x4_f32
//     (tile x 0.25*I chained over 4 K-slices) on the matrix pipe.
// ---------------------------------------------------------------------------

#define N_USER   50000
#define N_ITEM   50000
#define NNODE    (N_USER + N_ITEM)          // 100000
#define LATDIM   64
#define NEDGE    1250000
#define ND       (NNODE * LATDIM)           // 6,400,000 floats per view
#define KEEP_RATE 0.8f

#define EDGE_TILE 256                       // edges staged per buffer
#define STAGES    4                         // stages per block (1221 blocks:
                                            // wide WGP coverage for the
                                            // atomic-latency-bound phase)

typedef __attribute__((ext_vector_type(2))) float v2f;
typedef __attribute__((ext_vector_type(8))) float v8f;

// Hardware f32 atomic add (global_atomic_add_f32, no compare-exchange loop).
__device__ __forceinline__ void atomic_add_f32(float* p, float v) {
    unsafeAtomicAdd(p, v);
}

// Deterministic per-edge hash -> uniform [0,1).  (JAX PRNG not reproducible
// on-device; determinism across calls is what the harness requires.)
__device__ __forceinline__ float hash_u01(uint32_t x, uint32_t seed) {
    x ^= seed;
    x *= 0x9E3779B1u; x ^= x >> 16;
    x *= 0x85EBCA6Bu; x ^= x >> 13;
    x *= 0xC2B2AE35u; x ^= x >> 16;
    return (float)(x >> 8) * (1.0f / 16777216.0f);
}

// ---------------------------------------------------------------------------
// ini = concat(uEmbeds, iEmbeds); also seed all three output accumulators
// with the layer-0 term (reference adds `ini` into every view's sum).
// ---------------------------------------------------------------------------
__global__ void init_kernel(const float4* __restrict__ uE,
                            const float4* __restrict__ iE,
                            float4* __restrict__ ini,
                            float4* __restrict__ o0,
                            float4* __restrict__ o1,
                            float4* __restrict__ o2) {
    int i = blockIdx.x * blockDim.x + threadIdx.x;   // over ND/4
    if (i >= ND / 4) return;
    const int uCount = N_USER * LATDIM / 4;
    float4 v = (i < uCount) ? uE[i] : iE[i - uCount];
    ini[i] = v; o0[i] = v; o1[i] = v; o2[i] = v;
}

__global__ void mask_kernel(const float* __restrict__ val,
                            float* __restrict__ v1,
                            float* __restrict__ v2) {
    int e = blockIdx.x * blockDim.x + threadIdx.x;
    if (e >= NEDGE) return;
    float v = val[e];
    float a = hash_u01((uint32_t)e, 0x1234567u);
    float b = hash_u01((uint32_t)e, 0xABCDEF1u);
    v1[e] = (a < KEEP_RATE) ? v * (1.0f / KEEP_RATE) : 0.0f;
    v2[e] = (b < KEEP_RATE) ? v * (1.0f / KEEP_RATE) : 0.0f;
}

__global__ void zero_kernel(float4* __restrict__ y) {
    int i = blockIdx.x * blockDim.x + threadIdx.x;
    if (i < ND / 4) y[i] = make_float4(0.f, 0.f, 0.f, 0.f);
}

// dst += src, and zero the next layer's scatter target in the same pass
// (saves a dedicated 25.6MB zero pass per middle layer).
__global__ void accum_zero_kernel(float4* __restrict__ dst,
                                  const float4* __restrict__ src,
                                  float4* __restrict__ z) {
    int i = blockIdx.x * blockDim.x + threadIdx.x;
    if (i >= ND / 4) return;
    float4 d = dst[i], s = src[i];
    d.x += s.x; d.y += s.y; d.z += s.z; d.w += s.w;
    dst[i] = d;
    z[i] = make_float4(0.f, 0.f, 0.f, 0.f);
}

__global__ void accum_kernel(float4* __restrict__ dst, const float4* __restrict__ src) {
    int i = blockIdx.x * blockDim.x + threadIdx.x;
    if (i >= ND / 4) return;
    float4 d = dst[i], s = src[i];
    d.x += s.x; d.y += s.y; d.z += s.z; d.w += s.w;
    dst[i] = d;
}

// ---------------------------------------------------------------------------
// SpMM: y[row[e]] += w[e] * x[col[e]]  (y pre-zeroed).
// Edge triples streamed to LDS with async DMA, double buffered; ASYNCcnt
// paced with s_wait_asynccnt (3 outstanding ops per staged tile).
// ---------------------------------------------------------------------------
__global__ void __launch_bounds__(256)
spmm_kernel(const int* __restrict__ row, const int* __restrict__ col,
            const float* __restrict__ w, const float* __restrict__ x,
            float* __restrict__ y) {
    __shared__ int   sRow[2][EDGE_TILE];
    __shared__ int   sCol[2][EDGE_TILE];
    __shared__ float sVal[2][EDGE_TILE];

    const int tid = threadIdx.x;                              // 0..255
    const long long base = (long long)blockIdx.x * (EDGE_TILE * STAGES);

    auto prefetch = [&](int buf, long long e0) {
        long long e = e0 + tid;
        if (e > (long long)NEDGE - 1) e = NEDGE - 1;          // clamp (dup reads ok)
        unsigned lr = (unsigned)(uintptr_t)&sRow[buf][tid];   // low 32 bits of a
        unsigned lc = (unsigned)(uintptr_t)&sCol[buf][tid];   // generic LDS ptr
        unsigned lv = (unsigned)(uintptr_t)&sVal[buf][tid];   // == LDS offset
        unsigned long long gr = (unsigned long long)(uintptr_t)(row + e);
        unsigned long long gc = (unsigned long long)(uintptr_t)(col + e);
        unsigned long long gv = (unsigned long long)(uintptr_t)(w + e);
        asm volatile("global_load_async_to_lds_b32 %0, %1, off"
                     :: "v"(lr), "v"(gr) : "memory");
        asm volatile("global_load_async_to_lds_b32 %0, %1, off"
                     :: "v"(lc), "v"(gc) : "memory");
        asm volatile("global_load_async_to_lds_b32 %0, %1, off"
                     :: "v"(lv), "v"(gv) : "memory");
    };

    prefetch(0, base);
    for (int s = 0; s < STAGES; ++s) {
        const long long e0 = base + (long long)s * EDGE_TILE;
        if (e0 >= NEDGE) break;                               // block-uniform
        if (s + 1 < STAGES && e0 + EDGE_TILE < NEDGE) {
            prefetch((s + 1) & 1, e0 + EDGE_TILE);            // overlap next tile
            asm volatile("s_wait_asynccnt 0x3" ::: "memory"); // oldest tile done
        } else {
            asm volatile("s_wait_asynccnt 0x0" ::: "memory");
        }
        __syncthreads();

        const int buf = s & 1;
        const int grp = tid >> 4;          // 16 edges in flight per pass
        const int d0  = (tid & 15) * 4;    // lanes 0..15 -> contiguous 256B
        #pragma unroll 4
        for (int p = 0; p < EDGE_TILE / 16; ++p) {
            int el = p * 16 + grp;
            long long e = e0 + el;
            if (e < NEDGE) {
                float v = sVal[buf][el];
                if (v != 0.0f) {
                    int r = sRow[buf][el];
                    int c = sCol[buf][el];
                    const float4 xv = *(const float4*)(x + (long long)c * LATDIM + d0);
                    float* yp = y + (long long)r * LATDIM + d0;
                    atomic_add_f32(yp + 0, v * xv.x);
                    atomic_add_f32(yp + 1, v * xv.y);
                    atomic_add_f32(yp + 2, v * xv.z);
                    atomic_add_f32(yp + 3, v * xv.w);
                }
            }
        }
        __syncthreads();                   // before next overwrite of `buf`
    }
}

// ---------------------------------------------------------------------------
// In-place tile scale on the matrix pipe: for each 16x16 f32 tile T,
// D = sum_k A_k * B_k where A_k = T[:,4k:4k+4] and B_k = rows 4k..4k+3 of
// s*I(16), i.e. D = s*T.  One wave per tile; EXEC all-ones (grid sized exact).
// A 16x4 layout: lanes 0-15 hold row M=lane {K=0,1}, lanes 16-31 {K=2,3}.
// B 4x16 layout: VGPR v, lanes 0-15 = row K=v, lanes 16-31 = row K=v+2.
// C/D layout:    VGPR v, lanes 0-15 = (M=v, N=lane), lanes 16-31 = (M=v+8).
// ---------------------------------------------------------------------------
__global__ void __launch_bounds__(256)
wmma_scale_kernel(float* __restrict__ buf, int numTiles, float s) {
    const int wave = (int)((blockIdx.x * blockDim.x + threadIdx.x) >> 5);
    const int lane = threadIdx.x & 31;
    if (wave >= numTiles) return;          // wave-uniform
    float* t = buf + (long long)wave * 256;
    const int  m  = lane & 15;
    const bool hi = lane >= 16;
#if __has_builtin(__builtin_amdgcn_wmma_f32_16x16x4_f32)
    v8f c = {};
    #pragma unroll
    for (int k = 0; k < 4; ++k) {
        const int k0 = 4 * k + (hi ? 2 : 0);
        const int k1 = 4 * k + (hi ? 3 : 1);
        v2f a, b;
        a.x = t[m * 16 + k0];
        a.y = t[m * 16 + k1];
        b.x = (m == k0) ? s : 0.0f;        // row K of s*I: nonzero at N==K
        b.y = (m == k1) ? s : 0.0f;
        c = __builtin_amdgcn_wmma_f32_16x16x4_f32(
                /*neg_a=*/false, a, /*neg_b=*/false, b,
                /*c_mod=*/(short)0, c, /*reuse_a=*/false, /*reuse_b=*/false);
    }
    #pragma unroll
    for (int v = 0; v < 8; ++v)
        t[(v + (hi ? 8 : 0)) * 16 + m] = c[v];
#else
    // Fallback: plain VALU scale (same tile ownership).
    for (int j = lane; j < 256; j += 32) t[j] *= s;
#endif
}

// ---------------------------------------------------------------------------
extern "C" void kernel_launch(void* const* d_in, const int* in_sizes, int n_in,
                              void* d_out, int out_size, void* d_ws, size_t ws_size,
                              hipStream_t stream) {
    (void)in_sizes; (void)n_in; (void)out_size; (void)ws_size;

    const int*   row = (const int*)d_in[0];
    const int*   col = (const int*)d_in[1];
    const float* val = (const float*)d_in[2];
    const float* uE  = (const float*)d_in[3];
    const float* iE  = (const float*)d_in[4];

    float* out    = (float*)d_out;               // [main | view1 | view2], each ND
    float* main_o = out;
    float* v1_o   = out + (size_t)ND;
    float* v2_o   = out + (size_t)2 * ND;

    float* ws   = (float*)d_ws;                  // 3*ND + 2*NEDGE floats (~87MB)
    float* ini  = ws;
    float* bufA = ws + (size_t)ND;
    float* bufB = ws + (size_t)2 * ND;
    float* val1 = ws + (size_t)3 * ND;
    float* val2 = val1 + NEDGE;

    const dim3 blk(256);
    const int elemBlocks = (ND / 4 + 255) / 256;
    const int spmmBlocks = (NEDGE + EDGE_TILE * STAGES - 1) / (EDGE_TILE * STAGES);

    init_kernel<<<elemBlocks, blk, 0, stream>>>((const float4*)uE, (const float4*)iE,
        (float4*)ini, (float4*)main_o, (float4*)v1_o, (float4*)v2_o);
    mask_kernel<<<(NEDGE + 255) / 256, blk, 0, stream>>>(val, val1, val2);

    const float* wv[3]   = { val, val1, val2 };
    float*       outs[3] = { main_o, v1_o, v2_o };
    for (int view = 0; view < 3; ++view) {
        // layer schedule: ini->A, A->B, B->A ; fuse "zero next target" into
        // the accumulation pass of the preceding layer.
        zero_kernel<<<elemBlocks, blk, 0, stream>>>((float4*)bufA);

        spmm_kernel<<<spmmBlocks, blk, 0, stream>>>(row, col, wv[view], ini, bufA);
        accum_zero_kernel<<<elemBlocks, blk, 0, stream>>>((float4*)outs[view],
            (const float4*)bufA, (float4*)bufB);

        spmm_kernel<<<spmmBlocks, blk, 0, stream>>>(row, col, wv[view], bufA, bufB);
        accum_zero_kernel<<<elemBlocks, blk, 0, stream>>>((float4*)outs[view],
            (const float4*)bufB, (float4*)bufA);

        spmm_kernel<<<spmmBlocks, blk, 0, stream>>>(row, col, wv[view], bufB, bufA);
        accum_kernel<<<elemBlocks, blk, 0, stream>>>((float4*)outs[view],
            (const float4*)bufA);
    }

    // main = sum / (N_LAYERS + 1); ND/256 = 25000 tiles, 8 waves/block -> exact.
    wmma_scale_kernel<<<(ND / 256) * 32 / 256, blk, 0, stream>>>(main_o, ND / 256, 0.25f);
}